// RGCN_50268297232762
// MI455X (gfx1250) — compile-verified
//
#include <hip/hip_runtime.h>
#include <hip/hip_bf16.h>
#include <math.h>

// ---------------------------------------------------------------------------
// RGCN / bipartite-GAT forward for MI455X (gfx1250, wave32).
// GEMMs use V_WMMA_F32_16X16X4_F32 (fp32 end-to-end, matches fp32 reference),
// register-blocked 2x2: one wave owns a 32x32 C block (4 accumulators), so
// each k-step issues 4 WMMAs against 6 vector loads (A/B fragments reused
// across two WMMAs each).
// ---------------------------------------------------------------------------

typedef float v2f __attribute__((ext_vector_type(2)));
typedef float v8f __attribute__((ext_vector_type(8)));

#define NNODES 20000
#define NEDGES 200000
#define KDIM   256

#define WMMA_F32(a, b, c)                                                  \
  __builtin_amdgcn_wmma_f32_16x16x4_f32(false, (a), false, (b), (short)0,  \
                                        (c), false, false)

// ---------------------------------------------------------------------------
// GEMM: C[M,Nc] = A[M,K] @ W[K,Nc] (+ bias[Nc]), fp32 WMMA 16x16x4.
// One wave per 32x32 C block (2x2 tiles). M % 32 == 0, Nc % 32 == 0, K % 4 == 0.
// A frag (16x4): lane<16 -> M=lane, comps K=k+0,k+1 ; lane>=16 -> K=k+2,k+3.
// B frag (4x16): lane&15 = N column, lane-half selects K pair (mirrors A).
// C/D: VGPR j -> row m0 + 8*(lane>>4) + j, col n0 + (lane&15).
// ---------------------------------------------------------------------------
__global__ __launch_bounds__(256) void rgcn_gemm_wmma_f32(
    const float* __restrict__ A, const float* __restrict__ W,
    const float* __restrict__ bias, float* __restrict__ C,
    int M, int K, int Nc) {
  int nst   = Nc >> 5;  // 32-column strips
  int wave  = (int)((blockIdx.x * blockDim.x + threadIdx.x) >> 5);
  int total = (M >> 5) * nst;
  if (wave >= total) return;
  int lane = threadIdx.x & 31;
  int m0 = (wave / nst) << 5;
  int n0 = (wave % nst) << 5;
  int half = lane >> 4;
  int mr   = lane & 15;

  const float* arow0 = A + (size_t)(m0 + mr) * K + 2 * half;
  const float* arow1 = arow0 + (size_t)16 * K;
  const float* bcol0 = W + (size_t)(2 * half) * Nc + n0 + mr;
  const float* bcol1 = bcol0 + 16;

  v8f acc00 = {}, acc01 = {}, acc10 = {}, acc11 = {};
#pragma unroll 4
  for (int k = 0; k < K; k += 4) {
    v2f a0, a1, b0, b1;
    a0.x = arow0[k];
    a0.y = arow0[k + 1];
    a1.x = arow1[k];
    a1.y = arow1[k + 1];
    b0.x = bcol0[(size_t)k * Nc];
    b0.y = bcol0[(size_t)k * Nc + Nc];
    b1.x = bcol1[(size_t)k * Nc];
    b1.y = bcol1[(size_t)k * Nc + Nc];
    acc00 = WMMA_F32(a0, b0, acc00);
    acc01 = WMMA_F32(a0, b1, acc01);
    acc10 = WMMA_F32(a1, b0, acc10);
    acc11 = WMMA_F32(a1, b1, acc11);
  }

  float bv0 = bias ? bias[n0 + mr] : 0.0f;
  float bv1 = bias ? bias[n0 + 16 + mr] : 0.0f;
  float* c0 = C + (size_t)(m0 + 8 * half) * Nc + n0 + mr;
  float* c1 = c0 + (size_t)16 * Nc;
#pragma unroll
  for (int j = 0; j < 8; ++j) {
    c0[(size_t)j * Nc]      = acc00[j] + bv0;
    c0[(size_t)j * Nc + 16] = acc01[j] + bv1;
    c1[(size_t)j * Nc]      = acc10[j] + bv0;
    c1[(size_t)j * Nc + 16] = acc11[j] + bv1;
  }
}

// el[n,h] = sum_d f[n, h*D+d] * avec[h*D+d]
__global__ void rgcn_att_dot(const float* __restrict__ f,
                             const float* __restrict__ avec,
                             float* __restrict__ out, int NH, int H, int D) {
  int i = blockIdx.x * blockDim.x + threadIdx.x;
  if (i >= NH) return;
  int n = i / H, h = i - n * H;
  const float* fr = f + (size_t)n * H * D + (size_t)h * D;
  const float* ar = avec + (size_t)h * D;
  float s = 0.0f;
  for (int d = 0; d < D; ++d) s += fr[d] * ar[d];
  out[i] = s;
}

__global__ void rgcn_fill(float* __restrict__ p, float v, int n) {
  int i = blockIdx.x * blockDim.x + threadIdx.x;
  if (i < n) p[i] = v;
}

__global__ void rgcn_fill_bias(float* __restrict__ p,
                               const float* __restrict__ bias, int n, int F) {
  int i = blockIdx.x * blockDim.x + threadIdx.x;
  if (i < n) p[i] = bias[i % F];
}

__device__ __forceinline__ void atomicMaxF32(float* addr, float val) {
  if (val >= 0.0f)
    atomicMax((int*)addr, __float_as_int(val));
  else
    atomicMin((unsigned int*)addr, __float_as_uint(val));
}

// e = leaky_relu(el[src]+er[dst], 0.2); store; segment max into mbuf[dst]
__global__ void rgcn_edge_logits(const float* __restrict__ el,
                                 const float* __restrict__ er,
                                 const int* __restrict__ src,
                                 const int* __restrict__ dst,
                                 float* __restrict__ ebuf,
                                 float* __restrict__ mbuf, int EH, int H) {
  int i = blockIdx.x * blockDim.x + threadIdx.x;
  if (i >= EH) return;
  int e = i / H, h = i - e * H;
  float x = el[(size_t)src[e] * H + h] + er[(size_t)dst[e] * H + h];
  x = x > 0.0f ? x : 0.2f * x;
  ebuf[i] = x;
  atomicMaxF32(&mbuf[(size_t)dst[e] * H + h], x);
}

// w = exp(e - m[dst]); store; segment sum into sbuf[dst]
__global__ void rgcn_edge_exp(const float* __restrict__ mbuf,
                              const int* __restrict__ dst,
                              float* __restrict__ ebuf,
                              float* __restrict__ sbuf, int EH, int H) {
  int i = blockIdx.x * blockDim.x + threadIdx.x;
  if (i >= EH) return;
  int e = i / H, h = i - e * H;
  float w = __expf(ebuf[i] - mbuf[(size_t)dst[e] * H + h]);
  ebuf[i] = w;
  atomicAdd(&sbuf[(size_t)dst[e] * H + h], w);
}

// a = w / s[dst]
__global__ void rgcn_edge_norm(const float* __restrict__ sbuf,
                               const int* __restrict__ dst,
                               float* __restrict__ ebuf, int EH, int H) {
  int i = blockIdx.x * blockDim.x + threadIdx.x;
  if (i >= EH) return;
  int e = i / H, h = i - e * H;
  ebuf[i] = ebuf[i] / sbuf[(size_t)dst[e] * H + h];
}

// g[dst, f] += fs[src, f] * a[e, f/D]   (coalesced over f)
__global__ void rgcn_edge_aggregate(const float* __restrict__ fs,
                                    const float* __restrict__ ebuf,
                                    const int* __restrict__ src,
                                    const int* __restrict__ dst,
                                    float* __restrict__ g, int EF, int F,
                                    int D, int H) {
  int i = blockIdx.x * blockDim.x + threadIdx.x;
  if (i >= EF) return;
  int e = i / F, f = i - e * F;
  int h = f / D;
  float a = ebuf[(size_t)e * H + h];
  atomicAdd(&g[(size_t)dst[e] * F + f], fs[(size_t)src[e] * F + f] * a);
}

// Per-column sum / sum-of-squares; blockDim.x == F (256), 64 rows per block.
__global__ __launch_bounds__(256) void rgcn_bn_stats(
    const float* __restrict__ x, float* __restrict__ mu,
    float* __restrict__ m2, int Mrows, int F) {
  int c = threadIdx.x;
  int r0 = blockIdx.x * 64;
  int r1 = r0 + 64;
  if (r1 > Mrows) r1 = Mrows;
  float s = 0.0f, s2 = 0.0f;
  for (int r = r0; r < r1; ++r) {
    float v = x[(size_t)r * F + c];
    s += v;
    s2 += v * v;
  }
  atomicAdd(&mu[c], s);
  atomicAdd(&m2[c], s2);
}

// mu <- scale = g * rsqrt(var+eps) ; m2 <- shift = be - mean*scale
__global__ void rgcn_bn_finalize(float* __restrict__ mu, float* __restrict__ m2,
                                 const float* __restrict__ g,
                                 const float* __restrict__ be, int Mrows,
                                 int F) {
  int c = blockIdx.x * blockDim.x + threadIdx.x;
  if (c >= F) return;
  float inv = 1.0f / (float)Mrows;
  float m = mu[c] * inv;
  float v = m2[c] * inv - m * m;
  float sc = g[c] * rsqrtf(v + 1e-5f);
  mu[c] = sc;
  m2[c] = be[c] - m * sc;
}

// h = act( bn(gat) + bn(res) ) with folded per-column affines
__global__ void rgcn_update_act(const float* __restrict__ gat,
                                const float* __restrict__ res,
                                const float* __restrict__ sc1,
                                const float* __restrict__ sh1,
                                const float* __restrict__ sc2,
                                const float* __restrict__ sh2,
                                float* __restrict__ h, int n, int F,
                                int actType) {
  int i = blockIdx.x * blockDim.x + threadIdx.x;
  if (i >= n) return;
  int c = i % F;
  float v = sc1[c] * gat[i] + sh1[c] + sc2[c] * res[i] + sh2[c];
  h[i] = (actType == 0) ? (v > 0.0f ? v : 0.01f * v) : tanhf(v);
}

// ---------------------------------------------------------------------------
// Input index map: recursive insertion-order flatten of setup_inputs():
//   0:xA 1:xB | params: gat[8]{A2B{W,al,ar,b},B2A{W,al,ar,b}} -> 2..65
//   res[7]{A{W,b,g,be},B{W,b,g,be}} -> 66..121 | bn[7]{g,be} -> 122..135
//   bn2[7]{g,be} -> 136..149 | 150:sAB 151:dAB 152:sBA 153:dBA
// ---------------------------------------------------------------------------
#define IDX_XA 0
#define IDX_XB 1
#define GAT_W(i, dir) (2 + (i)*8 + (dir)*4 + 0)  // dir: 0=A2B, 1=B2A
#define GAT_AL(i, dir) (2 + (i)*8 + (dir)*4 + 1)
#define GAT_AR(i, dir) (2 + (i)*8 + (dir)*4 + 2)
#define GAT_B(i, dir) (2 + (i)*8 + (dir)*4 + 3)
#define RES_W(i, t) (66 + (i)*8 + (t)*4 + 0)  // t: 0=A, 1=B
#define RES_B(i, t) (66 + (i)*8 + (t)*4 + 1)
#define RES_G(i, t) (66 + (i)*8 + (t)*4 + 2)
#define RES_BE(i, t) (66 + (i)*8 + (t)*4 + 3)
#define BN_G(i) (122 + (i)*2)
#define BN_BE(i) (123 + (i)*2)
#define BN2_G(i) (136 + (i)*2)
#define BN2_BE(i) (137 + (i)*2)
#define IDX_SAB 150
#define IDX_DAB 151
#define IDX_SBA 152
#define IDX_DBA 153

static inline int cdiv(int a, int b) { return (a + b - 1) / b; }

extern "C" void kernel_launch(void* const* d_in, const int* in_sizes, int n_in,
                              void* d_out, int out_size, void* d_ws,
                              size_t ws_size, hipStream_t stream) {
  (void)in_sizes; (void)n_in; (void)out_size; (void)ws_size;

  static const int HEADS[8] = {8, 8, 4, 4, 4, 2, 2, 2};
  static const int DIMS[8]  = {32, 32, 64, 64, 64, 128, 128, 64};

  const int Nn = NNODES, Ee = NEDGES;
  const size_t NF = (size_t)Nn * 256;

  float* ws = (float*)d_ws;
  float* hA = ws + 0 * NF;
  float* hB = ws + 1 * NF;
  float* gA = ws + 2 * NF;
  float* gB = ws + 3 * NF;
  float* p0 = ws + 4 * NF;
  float* p1 = ws + 5 * NF;
  float* el = ws + 6 * NF;
  float* er = el + (size_t)Nn * 8;
  float* mb = er + (size_t)Nn * 8;
  float* sb = mb + (size_t)Nn * 8;
  float* eb = sb + (size_t)Nn * 8;          // [E, Hmax=8]
  float* st = eb + (size_t)Ee * 8;          // 4 BNs x (mu,m2) of 256 = 2048

  auto F32 = [&](int i) { return (const float*)d_in[i]; };
  const int* sAB = (const int*)d_in[IDX_SAB];
  const int* dAB = (const int*)d_in[IDX_DAB];
  const int* sBA = (const int*)d_in[IDX_SBA];
  const int* dBA = (const int*)d_in[IDX_DBA];

  hipMemcpyAsync(hA, d_in[IDX_XA], NF * sizeof(float),
                 hipMemcpyDeviceToDevice, stream);
  hipMemcpyAsync(hB, d_in[IDX_XB], NF * sizeof(float),
                 hipMemcpyDeviceToDevice, stream);

  for (int i = 0; i < 8; ++i) {
    const int H = HEADS[i], D = DIMS[i], F = H * D;  // F = 256, last layer 128
    const int NH = Nn * H, EH = Ee * H, EF = Ee * F;
    const int gemmWaves = (Nn / 32) * (F / 32);
    const int gemmBlocks = cdiv(gemmWaves * 32, 256);

    for (int dir = 0; dir < 2; ++dir) {
      // dir 0: B2A params, src in B (sBA) -> dst in A (dBA), produce gA
      // dir 1: A2B params, src in A (sAB) -> dst in B (dAB), produce gB
      const float* xs = (dir == 0) ? hB : hA;
      const float* xd = (dir == 0) ? hA : hB;
      const int* src = (dir == 0) ? sBA : sAB;
      const int* dst = (dir == 0) ? dBA : dAB;
      const int pdir = (dir == 0) ? 1 : 0;
      const float* Wg = F32(GAT_W(i, pdir));
      const float* al = F32(GAT_AL(i, pdir));
      const float* ar = F32(GAT_AR(i, pdir));
      const float* bg = F32(GAT_B(i, pdir));
      float* gout = (i == 7) ? ((float*)d_out + (size_t)dir * Nn * 128)
                             : ((dir == 0) ? gA : gB);

      // projections fs = xs @ W, fd = xd @ W
      rgcn_gemm_wmma_f32<<<gemmBlocks, 256, 0, stream>>>(xs, Wg, nullptr, p0,
                                                         Nn, KDIM, F);
      rgcn_gemm_wmma_f32<<<gemmBlocks, 256, 0, stream>>>(xd, Wg, nullptr, p1,
                                                         Nn, KDIM, F);
      // attention logits per node
      rgcn_att_dot<<<cdiv(NH, 256), 256, 0, stream>>>(p0, al, el, NH, H, D);
      rgcn_att_dot<<<cdiv(NH, 256), 256, 0, stream>>>(p1, ar, er, NH, H, D);
      // edge softmax
      rgcn_fill<<<cdiv(NH, 256), 256, 0, stream>>>(mb, -INFINITY, NH);
      rgcn_fill<<<cdiv(NH, 256), 256, 0, stream>>>(sb, 0.0f, NH);
      rgcn_edge_logits<<<cdiv(EH, 256), 256, 0, stream>>>(el, er, src, dst, eb,
                                                          mb, EH, H);
      rgcn_edge_exp<<<cdiv(EH, 256), 256, 0, stream>>>(mb, dst, eb, sb, EH, H);
      rgcn_edge_norm<<<cdiv(EH, 256), 256, 0, stream>>>(sb, dst, eb, EH, H);
      // aggregate: gout = bias + sum_e a * fs[src]
      rgcn_fill_bias<<<cdiv(Nn * F, 256), 256, 0, stream>>>(gout, bg, Nn * F, F);
      rgcn_edge_aggregate<<<cdiv(EF, 256), 256, 0, stream>>>(p0, eb, src, dst,
                                                             gout, EF, F, D, H);
    }

    if (i < 7) {
      // residuals: p0 = hA @ WresA + bA ; p1 = hB @ WresB + bB
      const int rb = cdiv((Nn / 32) * 8 * 32, 256);  // (Nn/32)*(256/32) waves
      rgcn_gemm_wmma_f32<<<rb, 256, 0, stream>>>(hA, F32(RES_W(i, 0)),
                                                 F32(RES_B(i, 0)), p0, Nn,
                                                 KDIM, 256);
      rgcn_gemm_wmma_f32<<<rb, 256, 0, stream>>>(hB, F32(RES_W(i, 1)),
                                                 F32(RES_B(i, 1)), p1, Nn,
                                                 KDIM, 256);
      // batchnorm stats (4 BNs: gat-A, res-A, gat-B, res-B)
      rgcn_fill<<<cdiv(2048, 256), 256, 0, stream>>>(st, 0.0f, 2048);
      const int sbk = cdiv(Nn, 64);
      rgcn_bn_stats<<<sbk, 256, 0, stream>>>(gA, st + 0, st + 256, Nn, 256);
      rgcn_bn_stats<<<sbk, 256, 0, stream>>>(p0, st + 512, st + 768, Nn, 256);
      rgcn_bn_stats<<<sbk, 256, 0, stream>>>(gB, st + 1024, st + 1280, Nn, 256);
      rgcn_bn_stats<<<sbk, 256, 0, stream>>>(p1, st + 1536, st + 1792, Nn, 256);
      rgcn_bn_finalize<<<1, 256, 0, stream>>>(st + 0, st + 256, F32(BN_G(i)),
                                              F32(BN_BE(i)), Nn, 256);
      rgcn_bn_finalize<<<1, 256, 0, stream>>>(st + 512, st + 768,
                                              F32(RES_G(i, 0)),
                                              F32(RES_BE(i, 0)), Nn, 256);
      rgcn_bn_finalize<<<1, 256, 0, stream>>>(st + 1024, st + 1280,
                                              F32(BN2_G(i)), F32(BN2_BE(i)),
                                              Nn, 256);
      rgcn_bn_finalize<<<1, 256, 0, stream>>>(st + 1536, st + 1792,
                                              F32(RES_G(i, 1)),
                                              F32(RES_BE(i, 1)), Nn, 256);
      // h = act( bn(gat) + bn(res) )
      const int act = (i <= 3) ? 0 : 1;
      rgcn_update_act<<<cdiv(Nn * 256, 256), 256, 0, stream>>>(
          gA, p0, st + 0, st + 256, st + 512, st + 768, hA, Nn * 256, 256, act);
      rgcn_update_act<<<cdiv(Nn * 256, 256), 256, 0, stream>>>(
          gB, p1, st + 1024, st + 1280, st + 1536, st + 1792, hB, Nn * 256,
          256, act);
    }
  }
}